// ts_std_14362370638092
// MI455X (gfx1250) — compile-verified
//
#include <hip/hip_runtime.h>

typedef __attribute__((ext_vector_type(2))) float v2f;
typedef __attribute__((ext_vector_type(8))) float v8f;

#define T_DIM   512
#define F_DIM   256
#define WIN     5
#define S_OUT   508            // T - WIN + 1 (stride 1 covers T-WIN already)
#define TILE_S  64             // window-starts per block
#define TILE_F  32             // features per block
#define TILE_T  (TILE_S + WIN - 1)   // 68 time rows staged in LDS
#define FPAD    40             // row pitch in floats: 160B -> 16B aligned rows, bank-safe

__global__ __launch_bounds__(256) void ts_std_kernel(const float* __restrict__ x,
                                                     float* __restrict__ out) {
  __shared__ float tile[TILE_T * FPAD];

  const int s0 = blockIdx.x * TILE_S;   // first window start handled by this block
  const int f0 = blockIdx.y * TILE_F;   // first feature
  const int b  = blockIdx.z;

  const unsigned long long xbase =
      (unsigned long long)(const void*)(x + (unsigned long long)b * T_DIM * F_DIM);
  // low 32 bits of a generic pointer to LDS == byte offset within the group segment
  const unsigned lds0 = (unsigned)(unsigned long long)(const void*)&tile[0];

  // ---------------- Phase 1: async global -> LDS (B128, 16B per lane) -----------
  // 68 rows x 32 floats = 544 x 16B transfers; 256 threads -> 3 guarded rounds.
  int i = threadIdx.x;
  #pragma unroll
  for (int it = 0; it < 3; ++it) {
    if (i < (TILE_T * TILE_F / 4)) {
      int t  = i >> 3;            // LDS row 0..67
      int f4 = (i & 7) << 2;      // float4 column within the 32-wide strip
      int tg = s0 + t;            // clamp tail rows (only feed masked-out s >= 508)
      if (tg > T_DIM - 1) tg = T_DIM - 1;
      unsigned goff = (unsigned)(((tg * F_DIM) + f0 + f4) * 4);
      unsigned loff = lds0 + (unsigned)((t * FPAD + f4) * 4);
      asm volatile("global_load_async_to_lds_b128 %0, %1, %2"
                   :
                   : "v"(loff), "v"(goff), "s"(xbase)
                   : "memory");
    }
    i += 256;
  }
  asm volatile("s_wait_asynccnt 0" ::: "memory");
  __syncthreads();

  // ---------------- Phase 2: banded WMMA reduction ------------------------------
  const int lane = threadIdx.x & 31;
  const int wave = threadIdx.x >> 5;
  const int n    = lane & 15;               // A: feature offset / B,D: start offset
  const int kb   = (lane < 16) ? 0 : 2;     // K pair held by this half-wave
  const int srel = (wave >> 1) * 16;        // wave's start sub-tile (0,16,32,48)
  const int fsub = (wave & 1) * 16;         // wave's feature sub-tile (0,16)

  // B operand: 0/1 band indicator, B[k][s] = 1 iff window s covers time 4c+k.
  v2f bmat[WIN];
  #pragma unroll
  for (int c = 0; c < WIN; ++c) {
    int u = 4 * c + kb - n;
    bmat[c].x = (u >= 0     && u <= WIN - 1)     ? 1.0f : 0.0f;
    bmat[c].y = (u + 1 >= 0 && u + 1 <= WIN - 1) ? 1.0f : 0.0f;
  }

  v8f s1 = {};   // Sum x
  v8f s2 = {};   // Sum x^2
  #pragma unroll
  for (int c = 0; c < WIN; ++c) {
    int trel = srel + 4 * c + kb;           // max 48+16+2 = 66 (+1 below) < 68
    float a0 = tile[trel * FPAD + fsub + n];
    float a1 = tile[(trel + 1) * FPAD + fsub + n];
    v2f a   = {a0, a1};
    v2f asq = {a0 * a0, a1 * a1};
    // 8 args: (neg_a, A, neg_b, B, c_mod, C, reuse_a, reuse_b)
    s1 = __builtin_amdgcn_wmma_f32_16x16x4_f32(false, a,   false, bmat[c],
                                               (short)0, s1, false, false);
    s2 = __builtin_amdgcn_wmma_f32_16x16x4_f32(false, asq, false, bmat[c],
                                               (short)0, s2, false, false);
  }

  // ---------------- Phase 3: var/std + coalesced transposed store ---------------
  // D layout: VGPR r -> M = r (+8 for lanes 16-31) = feature offset; N = lane%16 = s.
  const int s_idx = s0 + srel + n;
  const int mb    = (lane < 16) ? 0 : 8;
  if (s_idx < S_OUT) {
    float* ob = out + (unsigned long long)b * F_DIM * S_OUT + s_idx;
    #pragma unroll
    for (int r = 0; r < 8; ++r) {
      float S1  = s1[r];
      float S2  = s2[r];
      float var = (S2 - S1 * S1 * (1.0f / WIN)) * (1.0f / (WIN - 1));
      var = var > 0.0f ? var : 0.0f;        // also maps NaN-free negatives to 0
      int fo = f0 + fsub + mb + r;
      ob[(unsigned long long)fo * S_OUT] = sqrtf(var);
    }
  }
}

extern "C" void kernel_launch(void* const* d_in, const int* in_sizes, int n_in,
                              void* d_out, int out_size, void* d_ws, size_t ws_size,
                              hipStream_t stream) {
  const float* x = (const float*)d_in[0];
  float* out     = (float*)d_out;
  const int B    = in_sizes[0] / (T_DIM * F_DIM);   // 64

  dim3 grid((S_OUT + TILE_S - 1) / TILE_S,          // 8 start tiles
            F_DIM / TILE_F,                         // 8 feature tiles
            B);                                     // 64 batches
  ts_std_kernel<<<grid, dim3(256), 0, stream>>>(x, out);
}